// DGCNN_Adaptor_19387482374363
// MI455X (gfx1250) — compile-verified
//
#include <hip/hip_runtime.h>
#include <hip/hip_bf16.h>
#include <cstddef>

typedef __attribute__((ext_vector_type(2))) float v2f;
typedef __attribute__((ext_vector_type(8))) float v8f;

#define NPTS   4096
#define BATCH  8
#define KNN_K  20
#define NEG_BIG -3.0e38f

__device__ __forceinline__ float lrelu(float y) { return y > 0.f ? y : 0.2f * y; }

__device__ __forceinline__ void topk_insert(float (&val)[KNN_K], int (&id)[KNN_K],
                                            float s, int j) {
  if (s > val[KNN_K - 1] || (s == val[KNN_K - 1] && j < id[KNN_K - 1])) {
    val[KNN_K - 1] = s; id[KNN_K - 1] = j;
#pragma unroll
    for (int t = KNN_K - 1; t > 0; --t) {
      bool sw = (val[t] > val[t - 1]) ||
                (val[t] == val[t - 1] && id[t] < id[t - 1]);
      if (sw) {
        float tv = val[t]; val[t] = val[t - 1]; val[t - 1] = tv;
        int   ti = id[t];  id[t]  = id[t - 1];  id[t - 1]  = ti;
      }
    }
  }
}

// ---------------------------------------------------------------------------
// xx[b][j] = sum_c X[b][c][j]^2  (per-column squared norms for knn scores)
// ---------------------------------------------------------------------------
__global__ __launch_bounds__(256)
void xx_kernel(const float* __restrict__ X, int bstride, int C, int N,
               float* __restrict__ xx) {
  int gid = blockIdx.x * 256 + threadIdx.x;   // BATCH*N total
  int b = gid / N, j = gid % N;
  const float* Xb = X + (size_t)b * bstride;
  float s = 0.f;
  for (int c = 0; c < C; ++c) { float v = Xb[(size_t)c * N + j]; s = fmaf(v, v, s); }
  xx[gid] = s;
}

// ---------------------------------------------------------------------------
// WMMA KNN: one workgroup per 16-row query tile. Gram tiles G = Xq^T * Xj via
// V_WMMA_F32_16X16X4_F32 (A-fragments for the query rows preloaded once and
// reused across all 256 j-tiles). score = 2*G - ||xj||^2 (order-equivalent to
// reference pd per row). Scores staged in a 16x512 LDS strip per chunk; each
// half-wave scans one row into a per-lane register top-20; final 16-way merge.
// ---------------------------------------------------------------------------
template <int C>
__global__ __launch_bounds__(256)
void knn_wmma_kernel(const float* __restrict__ X, int bstride, int N,
                     const float* __restrict__ xx, int* __restrict__ idx_out) {
  constexpr int KS = (C + 3) / 4;     // wmma k-steps
  constexpr int CHUNK = 512;
  __shared__ float sbuf[10240];       // 40KB: strip[16][512] / merge lists
  __shared__ float xxs[CHUNK];
  __shared__ int   sheads[16][16];

  const int tid = threadIdx.x;
  const int w = tid >> 5, lane = tid & 31, lmod = lane & 15, lhi = lane >> 4;
  const int tilesPerBatch = N / 16;
  const int b  = blockIdx.x / tilesPerBatch;
  const int i0 = (blockIdx.x % tilesPerBatch) * 16;
  const float* Xb  = X + (size_t)b * bstride;
  const float* xxb = xx + (size_t)b * N;

  // Preload A fragments for query rows i0..i0+15 (reused for every j tile).
  v2f afr[KS];
#pragma unroll
  for (int ks = 0; ks < KS; ++ks) {
#pragma unroll
    for (int v = 0; v < 2; ++v) {
      int k = ks * 4 + v + 2 * lhi;
      afr[ks][v] = (k < C) ? Xb[(size_t)k * N + i0 + lmod] : 0.f;
    }
  }

  float val[KNN_K]; int id[KNN_K];
#pragma unroll
  for (int t = 0; t < KNN_K; ++t) { val[t] = NEG_BIG; id[t] = 0x7fffffff; }

  float (*strip)[CHUNK] = (float (*)[CHUNK])sbuf;

  for (int j0 = 0; j0 < N; j0 += CHUNK) {
    for (int t = tid; t < CHUNK; t += 256) xxs[t] = xxb[j0 + t];
    __syncthreads();   // xxs ready; previous chunk's scan finished

    for (int jt = w; jt < CHUNK / 16; jt += 8) {
      int jb = j0 + jt * 16;
      v8f acc = {0.f, 0.f, 0.f, 0.f, 0.f, 0.f, 0.f, 0.f};
#pragma unroll
      for (int ks = 0; ks < KS; ++ks) {
        v2f bfr;
#pragma unroll
        for (int v = 0; v < 2; ++v) {
          int k = ks * 4 + v + 2 * lhi;
          bfr[v] = (k < C) ? Xb[(size_t)k * N + jb + lmod] : 0.f;
        }
        acc = __builtin_amdgcn_wmma_f32_16x16x4_f32(false, afr[ks], false, bfr,
                                                    (short)0, acc, false, false);
      }
#pragma unroll
      for (int v = 0; v < 8; ++v) {
        int m  = v + 8 * lhi;
        int cc = jt * 16 + lmod;
        strip[m][cc] = 2.0f * acc[v] - xxs[cc];
      }
    }
    __syncthreads();   // strip ready

    { // scan: lanes 0-15 -> row 2w, lanes 16-31 -> row 2w+1
      int r = 2 * w + lhi;
      for (int q = 0; q < CHUNK / 16; ++q) {
        int cc = lmod + q * 16;
        topk_insert(val, id, strip[r][cc], j0 + cc);
      }
    }
    __syncthreads();   // scan finished before strip is overwritten
  }

  // Dump per-lane sorted lists into LDS (reuse strip space).
  {
    int r = 2 * w + lhi;
    float* mv = sbuf;                        // [16][16*20]
    int*   mi = (int*)(sbuf + 5120);         // [16][16*20]
#pragma unroll
    for (int t = 0; t < KNN_K; ++t) {
      mv[r * 320 + lmod * KNN_K + t] = val[t];
      mi[r * 320 + lmod * KNN_K + t] = id[t];
    }
  }
  __syncthreads();

  if (tid < 16) {   // 16-way sorted merge, one thread per query row
    int r = tid;
    for (int l = 0; l < 16; ++l) sheads[r][l] = l * KNN_K;
    const float* mv = sbuf;
    const int*   mi = (const int*)(sbuf + 5120);
    int* op = idx_out + ((size_t)b * N + i0 + r) * KNN_K;
    for (int t = 0; t < KNN_K; ++t) {
      float bestv = NEG_BIG; int besti = 0x7fffffff; int bestl = 0;
      for (int l = 0; l < 16; ++l) {
        int h = sheads[r][l];
        bool ok = h < l * KNN_K + KNN_K;
        float v = ok ? mv[r * 320 + h] : NEG_BIG;
        int  ii = ok ? mi[r * 320 + h] : 0x7fffffff;
        if (v > bestv || (v == bestv && ii < besti)) { bestv = v; besti = ii; bestl = l; }
      }
      sheads[r][bestl]++;
      op[t] = besti;
    }
  }
}

// ---------------------------------------------------------------------------
// Edge conv: one workgroup (8 waves) per point. Builds F = [diff; center]
// (2*CIN x 32, cols >=20 zero-padded) in LDS, then conv1 (Wa, K1R real K) and
// optionally conv2 (Wb, K=64), both via V_WMMA_F32_16X16X4_F32, LeakyReLU with
// scale/bias fused, then max over the 20 neighbor columns.
// ---------------------------------------------------------------------------
template <int CIN, int K1R, int K1P, bool TWO>
__global__ __launch_bounds__(256)
void edgeconv_kernel(const float* __restrict__ X, int xbstride,
                     const int* __restrict__ idx,
                     const float* __restrict__ Wa, const float* __restrict__ ga,
                     const float* __restrict__ ba,
                     const float* __restrict__ Wb, const float* __restrict__ gb,
                     const float* __restrict__ bb,
                     float* __restrict__ out, int obstride, int N) {
  __shared__ float F[K1P][32];
  __shared__ float H1[64][32];
  __shared__ float H2[TWO ? 64 : 1][32];
  __shared__ int nbr[KNN_K];

  const int tid = threadIdx.x;
  const int n = blockIdx.x % N;
  const int b = blockIdx.x / N;
  const float* Xb = X + (size_t)b * xbstride;

  if (tid < KNN_K) nbr[tid] = idx[((size_t)b * N + n) * KNN_K + tid];
  __syncthreads();

  for (int e = tid; e < K1P * 32; e += 256) {
    int r = e / 32, col = e % 32;
    float v = 0.f;
    if (col < KNN_K && r < 2 * CIN) {
      int c = (r < CIN) ? r : r - CIN;
      float xc = Xb[c * N + n];
      v = (r < CIN) ? (Xb[c * N + nbr[col]] - xc) : xc;
    }
    F[r][col] = v;
  }
  __syncthreads();

  const int w = tid >> 5, lane = tid & 31;
  const int mt = w >> 1, nt = w & 1;
  const int lmod = lane & 15, lhi = lane >> 4;

  { // conv1
    v8f acc = {0.f, 0.f, 0.f, 0.f, 0.f, 0.f, 0.f, 0.f};
#pragma unroll
    for (int ks = 0; ks < K1P / 4; ++ks) {
      v2f a, bfr;
#pragma unroll
      for (int v = 0; v < 2; ++v) {
        int k = ks * 4 + v + 2 * lhi;
        a[v]   = (k < K1R) ? Wa[(mt * 16 + lmod) * K1R + k] : 0.f;
        bfr[v] = F[k][nt * 16 + lmod];
      }
      acc = __builtin_amdgcn_wmma_f32_16x16x4_f32(false, a, false, bfr,
                                                  (short)0, acc, false, false);
    }
#pragma unroll
    for (int v = 0; v < 8; ++v) {
      int rrow = mt * 16 + v + 8 * lhi;
      int ccol = nt * 16 + lmod;
      H1[rrow][ccol] = lrelu(acc[v] * ga[rrow] + ba[rrow]);
    }
  }
  __syncthreads();

  if (TWO) { // conv2, K = 64
    v8f acc = {0.f, 0.f, 0.f, 0.f, 0.f, 0.f, 0.f, 0.f};
#pragma unroll
    for (int ks = 0; ks < 16; ++ks) {
      v2f a, bfr;
#pragma unroll
      for (int v = 0; v < 2; ++v) {
        int k = ks * 4 + v + 2 * lhi;
        a[v]   = Wb[(mt * 16 + lmod) * 64 + k];
        bfr[v] = H1[k][nt * 16 + lmod];
      }
      acc = __builtin_amdgcn_wmma_f32_16x16x4_f32(false, a, false, bfr,
                                                  (short)0, acc, false, false);
    }
#pragma unroll
    for (int v = 0; v < 8; ++v) {
      int rrow = mt * 16 + v + 8 * lhi;
      int ccol = nt * 16 + lmod;
      H2[rrow][ccol] = lrelu(acc[v] * gb[rrow] + bb[rrow]);
    }
    __syncthreads();
  }

  float (*HF)[32] = TWO ? (float (*)[32])H2 : (float (*)[32])H1;
  if (tid < 64) {
    float m = NEG_BIG;
    for (int col = 0; col < KNN_K; ++col) m = fmaxf(m, HF[tid][col]);
    out[(size_t)b * obstride + (size_t)tid * N + n] = m;
  }
}

// ---------------------------------------------------------------------------
// Fusion: per workgroup, 16 points; X = LReLU(gf*(Wf@feats)+bf) [1024x16],
// then per-row max over the 16 points -> partial maxima buffer.
// ---------------------------------------------------------------------------
__global__ __launch_bounds__(256)
void fusion_kernel(const float* __restrict__ feats, const float* __restrict__ Wf,
                   const float* __restrict__ gf, const float* __restrict__ bf_,
                   float* __restrict__ partials, int N) {
  __shared__ float F[192][16];
  const int tid = threadIdx.x;
  const int chunk = blockIdx.x % 256;
  const int b = blockIdx.x / 256;
  const int n0 = chunk * 16;
  const float* fb = feats + (size_t)b * 192 * N;

  for (int e = tid; e < 192 * 16; e += 256) {
    int r = e / 16, col = e % 16;
    F[r][col] = fb[(size_t)r * N + n0 + col];
  }
  __syncthreads();

  const int w = tid >> 5, lane = tid & 31, lmod = lane & 15, lhi = lane >> 4;
  float* pout = partials + ((size_t)b * 256 + chunk) * 1024;

  for (int mi = 0; mi < 8; ++mi) {
    int mt = w * 8 + mi;
    v8f acc = {0.f, 0.f, 0.f, 0.f, 0.f, 0.f, 0.f, 0.f};
    for (int ks = 0; ks < 48; ++ks) {
      v2f a, bfr;
#pragma unroll
      for (int v = 0; v < 2; ++v) {
        int k = ks * 4 + v + 2 * lhi;
        a[v]   = Wf[(size_t)(mt * 16 + lmod) * 192 + k];
        bfr[v] = F[k][lmod];
      }
      acc = __builtin_amdgcn_wmma_f32_16x16x4_f32(false, a, false, bfr,
                                                  (short)0, acc, false, false);
    }
#pragma unroll
    for (int v = 0; v < 8; ++v) {
      int rrow = mt * 16 + v + 8 * lhi;
      float y = lrelu(acc[v] * gf[rrow] + bf_[rrow]);
      for (int m = 1; m <= 8; m <<= 1) y = fmaxf(y, __shfl_xor(y, m, 32));
      if (lmod == 0) pout[rrow] = y;
    }
  }
}

__global__ __launch_bounds__(256)
void gmax_kernel(const float* __restrict__ partials, float* __restrict__ gmax) {
  int t = blockIdx.x * 256 + threadIdx.x;   // 8192 total
  int b = t / 1024, c = t % 1024;
  float m = NEG_BIG;
  for (int ch = 0; ch < 256; ++ch)
    m = fmaxf(m, partials[((size_t)b * 256 + ch) * 1024 + c]);
  gmax[t] = m;
}

// v1[b][m] = Wp1[m, 0:1024] . gmax[b]   (broadcast part of Wp1 hoisted out)
__global__ __launch_bounds__(256)
void v1_kernel(const float* __restrict__ Wp1, const float* __restrict__ gmax,
               float* __restrict__ v1) {
  int t = blockIdx.x * 256 + threadIdx.x;   // 4096 total
  int b = t / 512, m = t % 512;
  const float* g = gmax + (size_t)b * 1024;
  float s = 0.f;
  for (int c = 0; c < 1024; ++c) s = fmaf(Wp1[(size_t)m * 1216 + c], g[c], s);
  v1[t] = s;
}

// ---------------------------------------------------------------------------
// Head: per workgroup 16 points. X1 = LReLU(gp1*(v1 + Wp1[:,1024:]@F)+bp1),
// X2 = LReLU(gp2*(Wp2@X1)+bp2), out = Wp3@X2 + bp3.
// ---------------------------------------------------------------------------
__global__ __launch_bounds__(256)
void head_kernel(const float* __restrict__ feats,
                 const float* __restrict__ Wp1, const float* __restrict__ gp1,
                 const float* __restrict__ bp1,
                 const float* __restrict__ Wp2, const float* __restrict__ gp2,
                 const float* __restrict__ bp2,
                 const float* __restrict__ Wp3, const float* __restrict__ bp3,
                 const float* __restrict__ v1, float* __restrict__ out, int N) {
  __shared__ float F[192][16];
  __shared__ float X1[512][16];
  __shared__ float X2[256][16];
  const int tid = threadIdx.x;
  const int chunk = blockIdx.x % 256;
  const int b = blockIdx.x / 256;
  const int n0 = chunk * 16;
  const float* fb = feats + (size_t)b * 192 * N;

  for (int e = tid; e < 192 * 16; e += 256) {
    int r = e / 16, col = e % 16;
    F[r][col] = fb[(size_t)r * N + n0 + col];
  }
  __syncthreads();

  const int w = tid >> 5, lane = tid & 31, lmod = lane & 15, lhi = lane >> 4;
  const float* v1b = v1 + (size_t)b * 512;

  for (int s = 0; s < 4; ++s) {                 // 512 rows = 32 mtiles
    int mt = w + 8 * s;
    v8f acc;
#pragma unroll
    for (int v = 0; v < 8; ++v) acc[v] = v1b[mt * 16 + v + 8 * lhi];
    for (int ks = 0; ks < 48; ++ks) {           // K = 192
      v2f a, bfr;
#pragma unroll
      for (int v = 0; v < 2; ++v) {
        int k = ks * 4 + v + 2 * lhi;
        a[v]   = Wp1[(size_t)(mt * 16 + lmod) * 1216 + 1024 + k];
        bfr[v] = F[k][lmod];
      }
      acc = __builtin_amdgcn_wmma_f32_16x16x4_f32(false, a, false, bfr,
                                                  (short)0, acc, false, false);
    }
#pragma unroll
    for (int v = 0; v < 8; ++v) {
      int rrow = mt * 16 + v + 8 * lhi;
      X1[rrow][lmod] = lrelu(acc[v] * gp1[rrow] + bp1[rrow]);
    }
  }
  __syncthreads();

  for (int s = 0; s < 2; ++s) {                 // 256 rows = 16 mtiles
    int mt = w + 8 * s;
    v8f acc = {0.f, 0.f, 0.f, 0.f, 0.f, 0.f, 0.f, 0.f};
    for (int ks = 0; ks < 128; ++ks) {          // K = 512
      v2f a, bfr;
#pragma unroll
      for (int v = 0; v < 2; ++v) {
        int k = ks * 4 + v + 2 * lhi;
        a[v]   = Wp2[(size_t)(mt * 16 + lmod) * 512 + k];
        bfr[v] = X1[k][lmod];
      }
      acc = __builtin_amdgcn_wmma_f32_16x16x4_f32(false, a, false, bfr,
                                                  (short)0, acc, false, false);
    }
#pragma unroll
    for (int v = 0; v < 8; ++v) {
      int rrow = mt * 16 + v + 8 * lhi;
      X2[rrow][lmod] = lrelu(acc[v] * gp2[rrow] + bp2[rrow]);
    }
  }
  __syncthreads();

  if (tid < 13 * 16) {
    int o = tid / 16, col = tid % 16;
    float s = bp3[o];
    for (int c = 0; c < 256; ++c) s = fmaf(Wp3[o * 256 + c], X2[c][col], s);
    out[(size_t)b * 13 * N + (size_t)o * N + n0 + col] = s;
  }
}

// ---------------------------------------------------------------------------
extern "C" void kernel_launch(void* const* d_in, const int* in_sizes, int n_in,
                              void* d_out, int out_size, void* d_ws, size_t ws_size,
                              hipStream_t stream) {
  (void)in_sizes; (void)n_in; (void)out_size; (void)ws_size;
  const int N = NPTS;
  const float* inputs = (const float*)d_in[0];
  const float* W1 = (const float*)d_in[1];  const float* g1 = (const float*)d_in[2];  const float* b1 = (const float*)d_in[3];
  const float* W2 = (const float*)d_in[4];  const float* g2 = (const float*)d_in[5];  const float* b2 = (const float*)d_in[6];
  const float* W3 = (const float*)d_in[7];  const float* g3 = (const float*)d_in[8];  const float* b3 = (const float*)d_in[9];
  const float* W4 = (const float*)d_in[10]; const float* g4 = (const float*)d_in[11]; const float* b4 = (const float*)d_in[12];
  const float* W5 = (const float*)d_in[13]; const float* g5 = (const float*)d_in[14]; const float* b5 = (const float*)d_in[15];
  const float* Wf = (const float*)d_in[16]; const float* gf = (const float*)d_in[17]; const float* bf = (const float*)d_in[18];
  const float* Wp1 = (const float*)d_in[19]; const float* gp1 = (const float*)d_in[20]; const float* bp1 = (const float*)d_in[21];
  const float* Wp2 = (const float*)d_in[22]; const float* gp2 = (const float*)d_in[23]; const float* bp2 = (const float*)d_in[24];
  const float* Wp3 = (const float*)d_in[25]; const float* bp3 = (const float*)d_in[26];
  float* out = (float*)d_out;

  // workspace layout
  const size_t idx_sz = (size_t)BATCH * N * KNN_K;           // ints
  int* idx1 = (int*)d_ws;
  int* idx2 = idx1 + idx_sz;
  int* idx3 = idx2 + idx_sz;
  float* feats    = (float*)(idx3 + idx_sz);                 // [8][192][N]
  float* partials = feats + (size_t)BATCH * 192 * N;         // [8][256][1024]
  float* gmax     = partials + (size_t)BATCH * 256 * 1024;   // [8][1024]
  float* v1       = gmax + (size_t)BATCH * 1024;             // [8][512]
  float* xxbuf    = v1 + (size_t)BATCH * 512;                // [8][N]

  const int xx_blocks  = BATCH * N / 256;
  const int knn_blocks = BATCH * (N / 16);
  const int pt_blocks  = BATCH * N;

  // Stage 1: knn on xyz (first 3 of 9 channels), edgeconv1 -> feats ch 0..63
  xx_kernel<<<xx_blocks, 256, 0, stream>>>(inputs, 9 * N, 3, N, xxbuf);
  knn_wmma_kernel<3><<<knn_blocks, 256, 0, stream>>>(inputs, 9 * N, N, xxbuf, idx1);
  edgeconv_kernel<9, 18, 20, true><<<pt_blocks, 256, 0, stream>>>(
      inputs, 9 * N, idx1, W1, g1, b1, W2, g2, b2, feats, 192 * N, N);

  // Stage 2: knn on net1, edgeconv2 -> feats ch 64..127
  xx_kernel<<<xx_blocks, 256, 0, stream>>>(feats, 192 * N, 64, N, xxbuf);
  knn_wmma_kernel<64><<<knn_blocks, 256, 0, stream>>>(feats, 192 * N, N, xxbuf, idx2);
  edgeconv_kernel<64, 128, 128, true><<<pt_blocks, 256, 0, stream>>>(
      feats, 192 * N, idx2, W3, g3, b3, W4, g4, b4, feats + (size_t)64 * N, 192 * N, N);

  // Stage 3: knn on net2, edgeconv3 (single conv) -> feats ch 128..191
  xx_kernel<<<xx_blocks, 256, 0, stream>>>(feats + (size_t)64 * N, 192 * N, 64, N, xxbuf);
  knn_wmma_kernel<64><<<knn_blocks, 256, 0, stream>>>(feats + (size_t)64 * N, 192 * N, N, xxbuf, idx3);
  edgeconv_kernel<64, 128, 128, false><<<pt_blocks, 256, 0, stream>>>(
      feats + (size_t)64 * N, 192 * N, idx3, W5, g5, b5,
      nullptr, nullptr, nullptr, feats + (size_t)128 * N, 192 * N, N);

  // Stage 4: fusion max, v1 precompute, MLP head
  fusion_kernel<<<BATCH * 256, 256, 0, stream>>>(feats, Wf, gf, bf, partials, N);
  gmax_kernel<<<BATCH * 1024 / 256, 256, 0, stream>>>(partials, gmax);
  v1_kernel<<<BATCH * 512 / 256, 256, 0, stream>>>(Wp1, gmax, v1);
  head_kernel<<<BATCH * 256, 256, 0, stream>>>(
      feats, Wp1, gp1, bp1, Wp2, gp2, bp2, Wp3, bp3, v1, out, N);
}